// LSD_63178968924509
// MI455X (gfx1250) — compile-verified
//
#include <hip/hip_runtime.h>

typedef float v2f __attribute__((ext_vector_type(2)));
typedef float v4f __attribute__((ext_vector_type(4)));
typedef float v8f __attribute__((ext_vector_type(8)));
typedef unsigned int v4u __attribute__((ext_vector_type(4)));
typedef int v4i __attribute__((ext_vector_type(4)));
typedef int v8i __attribute__((ext_vector_type(8)));

#define NVOX (128 * 128 * 128)

// LDS float-index layout (single shared array; offsets are wave-relative):
//   [0..63]  : 64-entry Toeplitz Gaussian weight table (index d+32, zero outside band)
//   per wave : two TDM-filled input tile buffers (2176 floats each) + output staging (2112)
#define WT_OFF 0
#define INBUF_FLOATS 2176          // x: 16*132=2112; y/z: 128*17=2176 -> use max
#define OUTBUF_FLOATS (16 * 132)
#define WAVE_FLOATS (2 * INBUF_FLOATS + OUTBUF_FLOATS)
#define SMEM_FLOATS (64 + 2 * WAVE_FLOATS)

// ---------------------------------------------------------------------------
// Build the 10-channel per-label stack: mask, mask*coords (3), second moments (6).
// ---------------------------------------------------------------------------
__global__ __launch_bounds__(256) void lsd_stack(const int* __restrict__ seg,
                                                 float* __restrict__ buf, int label) {
  const int v = blockIdx.x * 256 + threadIdx.x;
  const float m = (seg[v] == label) ? 1.0f : 0.0f;
  const float c0 = (float)(v >> 14);
  const float c1 = (float)((v >> 7) & 127);
  const float c2 = (float)(v & 127);
  buf[0 * NVOX + v] = m;
  buf[1 * NVOX + v] = c0 * m;
  buf[2 * NVOX + v] = c1 * m;
  buf[3 * NVOX + v] = c2 * m;
  buf[4 * NVOX + v] = c0 * c0 * m;  // XS/YS order: 00,11,22,01,12,02
  buf[5 * NVOX + v] = c1 * c1 * m;
  buf[6 * NVOX + v] = c2 * c2 * m;
  buf[7 * NVOX + v] = c0 * c1 * m;
  buf[8 * NVOX + v] = c1 * c2 * m;
  buf[9 * NVOX + v] = c0 * c2 * m;
}

// ---------------------------------------------------------------------------
// Tensor Data Mover: DMA one 2-D tile (data_size=4B) from global into LDS with
// hardware padding (ISA cdna5 CDNA5 ISA D# groups 0/1; groups 2/3 zero
// for 2-D tiles). lds_off is the wave-relative LDS byte address.
// 6-arg builtin form (clang-23 / therock-10.0 headers).
// ---------------------------------------------------------------------------
__device__ __forceinline__ void tdm_load_tile(const float* gptr, unsigned lds_off,
                                              unsigned tile0, unsigned tile1,
                                              unsigned long long stride0,
                                              unsigned padbits) {
  const unsigned long long ga = (unsigned long long)(uintptr_t)gptr;
  v4u g0;
  g0[0] = 1u;                                             // count=1 (valid user D#)
  g0[1] = lds_off;                                        // lds_addr (bytes)
  g0[2] = (unsigned)(ga & 0xFFFFFFFFu);                   // global_addr[31:0]
  g0[3] = (unsigned)((ga >> 32) & 0x01FFFFFFu) | (2u << 30);  // addr[56:32] | type=2
  v8i g1;
  g1[0] = (int)((2u << 16) | (1u << 20) | padbits);       // data_size=4B, pad_enable, pad cfg
  g1[1] = (int)((tile0 & 0xFFFFu) << 16);                 // tensor_dim0 = tile0 (no OOB)
  g1[2] = (int)(((tile0 >> 16) & 0xFFFFu) | ((tile1 & 0xFFFFu) << 16));  // dim0 hi | dim1 lo
  g1[3] = (int)(((tile1 >> 16) & 0xFFFFu) | ((tile0 & 0xFFFFu) << 16));  // dim1 hi | tile_dim0
  g1[4] = (int)(tile1 & 0xFFFFu);                         // tile_dim1 | tile_dim2=0
  g1[5] = (int)(stride0 & 0xFFFFFFFFull);                 // tensor_dim0_stride lo
  g1[6] = (int)((stride0 >> 32) & 0xFFFFull);             // stride0 hi | stride1 lo = 0
  g1[7] = 0;                                              // stride1 hi = 0
  const v4i z4 = {};
  const v8i z8 = {};
  __builtin_amdgcn_tensor_load_to_lds(g0, g1, z4, z4, z8, 0);
}

// ---------------------------------------------------------------------------
// In-place separable Gaussian blur along one axis as a banded Toeplitz matmul:
// OUT(16x128) = IN(16x128) * T(128x128) via v_wmma_f32_16x16x4_f32.
// Element (line m, pos k) lives at base + m*T + k*S, base = ch*NVOX + (tl>>3)*Q + (tl&7)*P.
// Fully compile-time specialized per axis so the inner loop is straight-line:
//   x-pass: Q=16384 P=2048 T=128 S=1      TDM tile 128x16 pad(128->+4) -> line-major 132
//   y-pass: Q=16384 P=16   T=1   S=128    TDM tile 16x128 pad(16->+1)  -> pos-major 17
//   z-pass: Q=128   P=16   T=1   S=16384  TDM tile 16x128 pad(16->+1)  -> pos-major 17
// 2 waves/block, 4 tiles/wave, double-buffered TDM prefetch (depth 1).
// ---------------------------------------------------------------------------
template <int POSMAJ, int Q, int P, int T, int S>
__global__ __launch_bounds__(64) void lsd_blur(float* __restrict__ buf) {
  __shared__ __align__(16) float smem[SMEM_FLOATS];
  const int wave = threadIdx.x >> 5;
  const int lane = threadIdx.x & 31;
  const int nl = lane & 15;
  const int kl = (lane < 16) ? 0 : 2;

  // ---- Toeplitz Gaussian weight table, built once per block ----
  {
    float ksum = 0.0f;
#pragma unroll
    for (int dd = 0; dd <= 15; ++dd)
      ksum += (dd ? 2.0f : 1.0f) * __expf((float)(dd * dd) * (-1.0f / 50.0f));
    const int dd = (int)threadIdx.x - 32;  // blockDim==64 -> one entry per thread
    smem[WT_OFF + threadIdx.x] =
        (dd >= -15 && dd <= 15) ? __expf((float)(dd * dd) * (-1.0f / 50.0f)) / ksum : 0.0f;
  }
  __syncthreads();

  // ---- per-lane B fragments: 12 distinct 16x16 T-blocks (rel in {-1,0,1} x 4 K-steps)
  // B layout (4x16 f32, 2 VGPRs): comp c holds row K = 4*ks + c + (lane<16 ? 0 : 2).
  v2f B[12];
#pragma unroll
  for (int rel = 0; rel < 3; ++rel) {
#pragma unroll
    for (int ks = 0; ks < 4; ++ks) {
      const int d0 = (rel - 1) * 16 + nl - ks * 4 - kl;  // d = n - k, index d+32 in [0,63]
      B[rel * 4 + ks].x = smem[WT_OFF + d0 + 32];
      B[rel * 4 + ks].y = smem[WT_OFF + d0 + 31];
    }
  }

  const int in_base = 64 + wave * WAVE_FLOATS;
  float* O = smem + in_base + 2 * INBUF_FLOATS;

  constexpr unsigned tile0 = POSMAJ ? 16u : 128u;
  constexpr unsigned tile1 = POSMAJ ? 128u : 16u;
  constexpr unsigned long long str0 = POSMAJ ? (unsigned long long)S : 128ull;
  // pad: x-pass every 128 DW add 4 DW (codes 6,3); y/z every 16 DW add 1 DW (codes 3,0)
  constexpr unsigned padbits = POSMAJ ? ((3u << 22) | (0u << 25)) : ((6u << 22) | (3u << 25));

  const int tbase = (blockIdx.x * 2 + wave) * 4;  // 4 tiles per wave, 10*1024 tiles total

  size_t base[4];
#pragma unroll
  for (int i = 0; i < 4; ++i) {
    const int tile = tbase + i;
    const int ch = tile >> 10;
    const int tl = tile & 1023;
    base[i] = (size_t)ch * NVOX + (size_t)(tl >> 3) * Q + (size_t)(tl & 7) * P;
  }

  const unsigned lds_in0 =
      (unsigned)(unsigned long long)(uintptr_t)(smem + in_base) & 0xFFFFFFFFu;

  // prologue: DMA tile 0
  tdm_load_tile(buf + base[0], lds_in0, tile0, tile1, str0, padbits);

#pragma unroll
  for (int it = 0; it < 4; ++it) {
    if (it + 1 < 4) {
      // before TDM overwrites the other buffer, be sure our old LDS reads retired
      asm volatile("s_wait_dscnt 0x0" ::: "memory");
      tdm_load_tile(buf + base[it + 1],
                    lds_in0 + (unsigned)(((it + 1) & 1) * INBUF_FLOATS * 4),
                    tile0, tile1, str0, padbits);
      __builtin_amdgcn_s_wait_tensorcnt((short)1);  // current tile's DMA done
    } else {
      __builtin_amdgcn_s_wait_tensorcnt((short)0);
    }
    asm volatile("" ::: "memory");

    const float* IN = smem + in_base + (it & 1) * INBUF_FLOATS;
    const size_t tb = base[it];

    // ---- banded matmul: output block j needs only kb in [j-1, j+1] ----
#pragma unroll
    for (int j = 0; j < 8; ++j) {
      v8f acc = {};
      const int kb0 = (j > 0) ? j - 1 : 0;
      const int kb1 = (j < 7) ? j + 1 : 7;
#pragma unroll
      for (int kb = kb0; kb <= kb1; ++kb) {
#pragma unroll
        for (int ks = 0; ks < 4; ++ks) {
          const int kpos = kb * 16 + ks * 4 + kl;
          v2f a;
          if (POSMAJ) {  // pos-major stride 17: two conflict-free b32 reads
            a.x = IN[kpos * 17 + nl];
            a.y = IN[kpos * 17 + 17 + nl];
          } else {       // line-major stride 132: one conflict-free b64 read
            a = *(const v2f*)(IN + nl * 132 + kpos);
          }
          acc = __builtin_amdgcn_wmma_f32_16x16x4_f32(
              false, a, false, B[(j - kb + 1) * 4 + ks], (short)0, acc, false, false);
        }
      }
      // C/D layout: vgpr i -> (M = i or i+8, N = lane%16), position 16j+N.
      if (POSMAJ) {  // stage into LDS, coalesced writeback below
#pragma unroll
        for (int i = 0; i < 8; ++i) {
          const int m = (lane < 16) ? i : (i + 8);
          O[m * 132 + j * 16 + nl] = acc[i];
        }
      } else {       // x pass: N contiguous in memory -> direct coalesced stores
        const size_t ob = tb + (size_t)(j * 16 + nl) * S;
#pragma unroll
        for (int i = 0; i < 8; ++i) {
          const int m = (lane < 16) ? i : (i + 8);
          buf[ob + (size_t)m * T] = acc[i];
        }
      }
    }

    if (POSMAJ) {  // y/z passes: write back coalesced via LDS transpose
      const int m4 = (lane & 3) * 4;
      const int bp = lane >> 2;
#pragma unroll
      for (int i = 0; i < 16; ++i) {
        const int pos = i * 8 + bp;
        v4f g;
        g.x = O[(m4 + 0) * 132 + pos];
        g.y = O[(m4 + 1) * 132 + pos];
        g.z = O[(m4 + 2) * 132 + pos];
        g.w = O[(m4 + 3) * 132 + pos];
        *(v4f*)(buf + tb + m4 + (size_t)pos * S) = g;
      }
    }
  }
}

// ---------------------------------------------------------------------------
// Combine blurred moments into descriptors, masked accumulate into d_out.
// ---------------------------------------------------------------------------
__global__ __launch_bounds__(256) void lsd_combine(const int* __restrict__ seg,
                                                   const float* __restrict__ agg,
                                                   float* __restrict__ out,
                                                   int label, int first) {
  const int v = blockIdx.x * 256 + threadIdx.x;
  const float m = (seg[v] == label) ? 1.0f : 0.0f;
  const float c0 = (float)(v >> 14);
  const float c1 = (float)((v >> 7) & 127);
  const float c2 = (float)(v & 127);
  const float mass = agg[v];
  const float inv = 1.0f / ((mass > 0.0f) ? mass : 1.0f);
  const float o0 = agg[1 * NVOX + v] * inv;
  const float o1 = agg[2 * NVOX + v] * inv;
  const float o2 = agg[3 * NVOX + v] * inv;
  const float is2 = 1.0f / 25.0f;  // 1/sigma^2
  float d[10];
  d[0] = (o0 - c0) * 0.1f + 0.5f;  // (com-coord)/sigma*0.5 + 0.5
  d[1] = (o1 - c1) * 0.1f + 0.5f;
  d[2] = (o2 - c2) * 0.1f + 0.5f;
  d[3] = (agg[4 * NVOX + v] * inv - o0 * o0) * is2;
  d[4] = (agg[5 * NVOX + v] * inv - o1 * o1) * is2;
  d[5] = (agg[6 * NVOX + v] * inv - o2 * o2) * is2;
  d[6] = (agg[7 * NVOX + v] * inv - o0 * o1) * is2;
  d[7] = (agg[8 * NVOX + v] * inv - o1 * o2) * is2;
  d[8] = (agg[9 * NVOX + v] * inv - o2 * o0) * is2;
  d[9] = mass;
#pragma unroll
  for (int ch = 0; ch < 10; ++ch) {
    const float val = d[ch] * m;  // mask^2 == mask for binary mask
    if (first)
      out[ch * NVOX + v] = val;
    else
      out[ch * NVOX + v] += val;
  }
}

__global__ __launch_bounds__(256) void lsd_clip(float* __restrict__ out) {
  const int v = blockIdx.x * 256 + threadIdx.x;
  out[v] = fminf(fmaxf(out[v], 0.0f), 1.0f);
}

// ---------------------------------------------------------------------------
extern "C" void kernel_launch(void* const* d_in, const int* in_sizes, int n_in,
                              void* d_out, int out_size, void* d_ws, size_t ws_size,
                              hipStream_t stream) {
  const int* seg = (const int*)d_in[0];  // segmentation int32 (coords recomputed on device)
  float* out = (float*)d_out;            // 10 x 128^3 f32
  float* buf = (float*)d_ws;             // 10 x 128^3 f32 = 80 MB (L2-resident)

  const int vox_blocks = NVOX / 256;             // 8192
  const int blur_blocks = (10 * 1024) / (2 * 4); // 2 waves/block x 4 tiles/wave = 1280

  for (int label = 1; label <= 8; ++label) {
    lsd_stack<<<vox_blocks, 256, 0, stream>>>(seg, buf, label);
    // in-place separable blur, one TDM+WMMA pass per axis (compile-time specialized)
    lsd_blur<0, 16384, 2048, 128, 1><<<blur_blocks, 64, 0, stream>>>(buf);   // x
    lsd_blur<1, 16384, 16, 1, 128><<<blur_blocks, 64, 0, stream>>>(buf);     // y
    lsd_blur<1, 128, 16, 1, 16384><<<blur_blocks, 64, 0, stream>>>(buf);     // z
    lsd_combine<<<vox_blocks, 256, 0, stream>>>(seg, buf, out, label, label == 1);
  }
  lsd_clip<<<(10 * NVOX) / 256, 256, 0, stream>>>(out);
}